// RGCNaggregator_33526514713101
// MI455X (gfx1250) — compile-verified
//
#include <hip/hip_runtime.h>
#include <hip/hip_bf16.h>

typedef float v2f __attribute__((ext_vector_type(2)));
typedef float v4f __attribute__((ext_vector_type(4)));
typedef float v8f __attribute__((ext_vector_type(8)));

#define HDIM 256
#define SENTINEL 0x7FFFFFFF

// ---------------- init: map=INT_MAX, agg_compact=0, deg=0 -------------------
__global__ void init_ws(int* __restrict__ map, float* __restrict__ aggc,
                        float* __restrict__ deg, int N, int P) {
    int i = blockIdx.x * blockDim.x + threadIdx.x;
    if (i < N) map[i] = SENTINEL;
    if (i < P * HDIM) aggc[i] = 0.0f;
    if (i < P) deg[i] = 0.0f;
}

// ------- build node -> representative-position map + per-position mask ------
__global__ void build_map(const int* __restrict__ target_s, const int* __restrict__ len_s,
                          int* __restrict__ map, float* __restrict__ mask, int P, int T) {
    int p = blockIdx.x * blockDim.x + threadIdx.x;
    if (p < P) {
        atomicMin(&map[target_s[p]], p);
        int b_ = p / T, t_ = p % T;              // division only here, once
        mask[p] = (t_ < len_s[b_]) ? 1.0f : 0.0f;
    }
}

// ---------------- edge pass: filtered gather + scatter-add ------------------
// one wave (32 lanes) per edge, grid-stride; ~98% of edges exit after 8B read
__global__ void edge_agg(const int* __restrict__ src, const int* __restrict__ dst,
                         const int* __restrict__ erel, const int* __restrict__ node_ids,
                         const float* __restrict__ ent, const float* __restrict__ rel,
                         const int* __restrict__ map, float* __restrict__ aggc,
                         float* __restrict__ deg, int E) {
    int wave   = (blockIdx.x * blockDim.x + threadIdx.x) >> 5;
    int lane   = threadIdx.x & 31;
    int nwaves = (gridDim.x * blockDim.x) >> 5;
    for (int e = wave; e < E; e += nwaves) {
        int slot = map[dst[e]];
        if (slot == SENTINEL) continue;
        long nid = node_ids[src[e]];
        long rr  = erel[e];
        // lane owns 8 consecutive floats -> two b128 loads per source row
        const v4f* hs = (const v4f*)(ent + nid * HDIM + lane * 8);
        const v4f* ev = (const v4f*)(rel + rr  * HDIM + lane * 8);
        float* ac = aggc + (long)slot * HDIM + lane * 8;
        v4f s0 = hs[0] + ev[0];
        v4f s1 = hs[1] + ev[1];
#pragma unroll
        for (int j = 0; j < 4; ++j) {
            atomicAdd(&ac[j],     s0[j]);
            atomicAdd(&ac[4 + j], s1[j]);
        }
        if (lane == 0) atomicAdd(&deg[slot], 1.0f);
    }
}

// ---------------- stage X = [agg/deg | h] as [P, 512] -----------------------
__global__ void stage_x(const int* __restrict__ target_s, const int* __restrict__ map,
                        const int* __restrict__ node_ids, const float* __restrict__ ent,
                        const float* __restrict__ aggc, const float* __restrict__ deg,
                        float* __restrict__ X) {
    int p = blockIdx.x;
    int c = threadIdx.x;                 // 0..255
    int node = target_s[p];
    int slot = map[node];
    float dv = fmaxf(deg[slot], 1.0f);
    long nid = node_ids[node];
    X[(long)p * (2 * HDIM) + c]        = aggc[(long)slot * HDIM + c] / dv;
    X[(long)p * (2 * HDIM) + HDIM + c] = ent[nid * HDIM + c];
}

// ---------------- stage Wcat = [W_msg ; W_self] as [512, 256] ---------------
__global__ void stage_w(const float* __restrict__ Wmsg, const float* __restrict__ Wself,
                        float* __restrict__ W) {
    int i = blockIdx.x * blockDim.x + threadIdx.x;  // over 512*256
    int k = i / HDIM, n = i % HDIM;
    W[i] = (k < HDIM) ? Wmsg[k * HDIM + n] : Wself[(k - HDIM) * HDIM + n];
}

// ---------------- fp32 WMMA GEMM: out[:, :256] = relu(X @ Wcat) * mask ------
// one wave per 16x16 tile; K = 512 in steps of 4 (V_WMMA_F32_16X16X4_F32)
__global__ void gemm_wmma(const float* __restrict__ X,    // [P, 512]
                          const float* __restrict__ W,    // [512, 256]
                          const float* __restrict__ mask, // [P]
                          float* __restrict__ out,        // [P, 512]
                          int P) {
    const int K = 2 * HDIM;
    int wave = (blockIdx.x * blockDim.x + threadIdx.x) >> 5;
    int lane = threadIdx.x & 31;
    int Ntiles = HDIM / 16;
    int Mtiles = P / 16;
    if (wave >= Mtiles * Ntiles) return;
    int tm = wave / Ntiles, tn = wave % Ntiles;
    int m0 = tm * 16, n0 = tn * 16;
    int half = lane >> 4;                 // selects K pair {0,1} vs {2,3}
    int mn   = lane & 15;                 // M for A-frag, N for B-frag

    v8f acc = {};
    const float* arow = X + (long)(m0 + mn) * K;
    const float* bcol = W + (n0 + mn);
    for (int k0 = 0; k0 < K; k0 += 4) {
        int ka = k0 + half * 2;
        v2f a = *(const v2f*)(arow + ka);           // A[m][ka], A[m][ka+1]
        v2f b;
        b.x = bcol[(long)(ka + 0) * HDIM];          // W[ka][n]
        b.y = bcol[(long)(ka + 1) * HDIM];          // W[ka+1][n]
        acc = __builtin_amdgcn_wmma_f32_16x16x4_f32(
            false, a, false, b, (short)0, acc, false, false);
    }
    // C/D layout: VGPR r, lane-half h -> row m0 + h*8 + r, col n0 + mn
#pragma unroll
    for (int r = 0; r < 8; ++r) {
        int row = m0 + half * 8 + r;        // flat position p = b*T + t
        int col = n0 + mn;
        float v = acc[r];
        v = v > 0.0f ? v : 0.0f;
        out[(long)row * (2 * HDIM) + col] = v * mask[row];
    }
}

// ---------------- out[:, 256:] = global_emb[g_idx] * mask -------------------
__global__ void fill_global(const float* __restrict__ gemb, const int* __restrict__ g_idx,
                            const float* __restrict__ mask, float* __restrict__ out) {
    int p = blockIdx.x;
    int c = threadIdx.x;
    long gi = g_idx[p];
    out[(long)p * (2 * HDIM) + HDIM + c] = gemb[gi * HDIM + c] * mask[p];
}

extern "C" void kernel_launch(void* const* d_in, const int* in_sizes, int n_in,
                              void* d_out, int out_size, void* d_ws, size_t ws_size,
                              hipStream_t stream) {
    const float* ent    = (const float*)d_in[0];
    const float* rel    = (const float*)d_in[1];
    const float* gemb   = (const float*)d_in[2];
    const float* Wmsg   = (const float*)d_in[3];
    const float* Wself  = (const float*)d_in[4];
    const int*   nids   = (const int*)d_in[5];
    const int*   src    = (const int*)d_in[6];
    const int*   dst    = (const int*)d_in[7];
    const int*   erel   = (const int*)d_in[8];
    const int*   tgt    = (const int*)d_in[9];
    const int*   g_idx  = (const int*)d_in[10];
    const int*   len_s  = (const int*)d_in[11];
    float* out = (float*)d_out;

    int N = in_sizes[5];          // 50000 nodes
    int E = in_sizes[6];          // 320000 edges
    int P = in_sizes[9];          // B*T = 1024 positions
    int B = in_sizes[11];         // 64
    int T = P / B;                // 16

    // workspace carve-up (256-byte aligned slices)
    char* ws = (char*)d_ws;
    size_t off = 0;
    auto take = [&](size_t bytes) { void* p = ws + off; off = (off + bytes + 255) & ~(size_t)255; return p; };
    int*   map  = (int*)take((size_t)N * 4);
    float* deg  = (float*)take((size_t)P * 4);
    float* msk  = (float*)take((size_t)P * 4);
    float* aggc = (float*)take((size_t)P * HDIM * 4);
    float* X    = (float*)take((size_t)P * 2 * HDIM * 4);
    float* Wcat = (float*)take((size_t)2 * HDIM * HDIM * 4);

    int initN = P * HDIM > N ? P * HDIM : N;
    init_ws<<<(initN + 255) / 256, 256, 0, stream>>>(map, aggc, deg, N, P);
    build_map<<<(P + 255) / 256, 256, 0, stream>>>(tgt, len_s, map, msk, P, T);
    edge_agg<<<1024, 256, 0, stream>>>(src, dst, erel, nids, ent, rel, map, aggc, deg, E);
    stage_x<<<P, HDIM, 0, stream>>>(tgt, map, nids, ent, aggc, deg, X);
    stage_w<<<(2 * HDIM * HDIM) / 256, 256, 0, stream>>>(Wmsg, Wself, Wcat);

    int waves = (P / 16) * (HDIM / 16);           // 1024 tiles, 1 wave each
    int threads = 128;                            // 4 waves / block
    int blocks = (waves * 32 + threads - 1) / threads;
    gemm_wmma<<<blocks, threads, 0, stream>>>(X, Wcat, msk, out, P);
    fill_global<<<P, HDIM, 0, stream>>>(gemb, g_idx, msk, out);
}